// DLMLayer_15564961481115
// MI455X (gfx1250) — compile-verified
//
#include <hip/hip_runtime.h>
#include <hip/hip_bf16.h>

// B=16, N=32, C=8, O=16
#define BB_ 16
#define N_  32
#define C_  8
#define O_  16

typedef __attribute__((ext_vector_type(2))) float v2f;
typedef __attribute__((ext_vector_type(8))) float v8f;

// sigmoid with fast hardware reciprocal (v_rcp_f32, ~1 ulp) + v_exp_f32
__device__ __forceinline__ float sigmoidf_(float v) {
    return __builtin_amdgcn_rcpf(1.0f + __expf(-v));
}

// D = A(16x4,f32) * B(4x16,f32) + C(16x16,f32), full-wave op (wave32)
__device__ __forceinline__ v8f wmma_k4(v2f a, v2f b, v8f c) {
#if __has_builtin(__builtin_amdgcn_wmma_f32_16x16x4_f32)
    return __builtin_amdgcn_wmma_f32_16x16x4_f32(
        /*neg_a=*/false, a, /*neg_b=*/false, b,
        /*c_mod=*/(short)0, c, /*reuse_a=*/false, /*reuse_b=*/false);
#else
    c[0] += a[0] * b[0] + a[1] * b[1];   // placeholder; never expected on gfx1250
    return c;
#endif
}

// ---------------- reduction kernels (fill workspace) ----------------

// red1[b*16 + c2]: c2<8 -> max_i x1[b,i,c2] ; c2>=8 -> min_i x1[b,i,c2-8]
__global__ void red1_kernel(const float* __restrict__ x1, float* __restrict__ red1) {
    int t  = threadIdx.x;          // 256 threads
    int c2 = t & 15;
    int b  = t >> 4;
    bool ismin = (c2 >= 8);
    int c = c2 & 7;
    float acc = ismin ? 3.402823466e38f : -3.402823466e38f;
    for (int i = 0; i < N_; ++i) {
        float v = x1[(b * N_ + i) * C_ + c];
        acc = ismin ? fminf(acc, v) : fmaxf(acc, v);
    }
    red1[t] = acc;
}

// red2[(b*N+i)*16 + c2]: reduce x2[b,i,j,c] over j, diagonal j==i -> 0 (max) / 1 (min)
__global__ void red2_kernel(const float* __restrict__ x2, float* __restrict__ red2) {
    int t  = blockIdx.x * blockDim.x + threadIdx.x;   // 8192 threads
    int c2 = t & 15;
    int i  = (t >> 4) & (N_ - 1);
    int b  = t >> 9;
    bool ismin = (c2 >= 8);
    int c = c2 & 7;
    float acc = ismin ? 3.402823466e38f : -3.402823466e38f;
    for (int j = 0; j < N_; ++j) {
        float v = x2[((b * N_ + i) * N_ + j) * C_ + c];
        if (j == i) v = ismin ? 1.0f : 0.0f;
        acc = ismin ? fminf(acc, v) : fmaxf(acc, v);
    }
    red2[t] = acc;
}

// red3[((b*N+i)*N+k)*16 + c2]: reduce x3[b,i,k,l,c] over l, diagonal l==k -> 0/1
__global__ void red3_kernel(const float* __restrict__ x3, float* __restrict__ red3) {
    int t  = blockIdx.x * blockDim.x + threadIdx.x;   // 262144 threads
    int c2 = t & 15;
    int k  = (t >> 4) & (N_ - 1);
    int i  = (t >> 9) & (N_ - 1);
    int b  = t >> 14;
    bool ismin = (c2 >= 8);
    int c = c2 & 7;
    float acc = ismin ? 3.402823466e38f : -3.402823466e38f;
    const float* base = x3 + (size_t)((b * N_ + i) * N_ + k) * (N_ * C_) + c;
    for (int l = 0; l < N_; ++l) {
        float v = base[l * C_];
        if (l == k) v = ismin ? 1.0f : 0.0f;
        acc = ismin ? fminf(acc, v) : fmaxf(acc, v);
    }
    red3[t] = acc;
}

// ---------------- WMMA GEMM kernels ----------------
// One wave per 16x16 output tile. Lane decomposition (f32 WMMA layouts):
//   half = lane>>4 ; mn = lane&15
//   A frag: row M = mn, K-in-chunk = 2*half + e  (e = vector element 0/1)
//   B frag: col N = mn, same K-in-chunk
//   C/D:    element d -> row d + 8*half, col mn
// W + bias staged in LDS once per block (8 waves reuse -> ds_load path).

// out0: rows = batch b (16 rows exactly), K=24 (x0:8, red1:16). Single wave.
__global__ void out0_kernel(const float* __restrict__ x0, const float* __restrict__ red1,
                            const float* __restrict__ W0, const float* __restrict__ bias0,
                            float* __restrict__ out0) {
    int lane = threadIdx.x & 31;
    int half = lane >> 4;
    int mn   = lane & 15;
    v8f acc = {0.f,0.f,0.f,0.f,0.f,0.f,0.f,0.f};
#pragma unroll
    for (int kk = 0; kk < 6; ++kk) {
        int q0 = 4 * kk + (half ? 2 : 0);
        v2f a, bm;
#pragma unroll
        for (int e = 0; e < 2; ++e) {
            int q = q0 + e;
            float av;
            if (q < 8) av = x0[mn * C_ + q];
            else       av = red1[mn * 16 + (q - 8)];
            a[e]  = av;
            bm[e] = W0[q * O_ + mn];
        }
        acc = wmma_k4(a, bm, acc);
    }
    float bv = bias0[mn];
#pragma unroll
    for (int d = 0; d < 8; ++d) {
        int m = d + 8 * half;
        out0[m * O_ + mn] = sigmoidf_(acc[d] + bv);
    }
}

// out1: rows = (b, i) flat (512), K=32 (x0:8, x1:8, red2:16)
__global__ void out1_kernel(const float* __restrict__ x0, const float* __restrict__ x1,
                            const float* __restrict__ red2,
                            const float* __restrict__ W1, const float* __restrict__ bias1,
                            float* __restrict__ out1) {
    __shared__ float sW[32 * O_ + O_];
    for (int idx = threadIdx.x; idx < 32 * O_ + O_; idx += blockDim.x)
        sW[idx] = (idx < 32 * O_) ? W1[idx] : bias1[idx - 32 * O_];
    __syncthreads();

    int wave = (int)(blockIdx.x * blockDim.x + threadIdx.x) >> 5;  // 0..31
    int lane = threadIdx.x & 31;
    int half = lane >> 4;
    int mn   = lane & 15;
    int it = wave & 1;          // i-tile (N/16 = 2)
    int b  = wave >> 1;
    int i  = it * 16 + mn;      // this lane's A-row i
    v8f acc = {0.f,0.f,0.f,0.f,0.f,0.f,0.f,0.f};
#pragma unroll
    for (int kk = 0; kk < 8; ++kk) {
        int q0 = 4 * kk + (half ? 2 : 0);
        v2f a, bm;
#pragma unroll
        for (int e = 0; e < 2; ++e) {
            int q = q0 + e;
            float av;
            if (q < 8)       av = x0[b * C_ + q];
            else if (q < 16) av = x1[(b * N_ + i) * C_ + (q - 8)];
            else             av = red2[(b * N_ + i) * 16 + (q - 16)];
            a[e]  = av;
            bm[e] = sW[q * O_ + mn];
        }
        acc = wmma_k4(a, bm, acc);
    }
    float bv = sW[32 * O_ + mn];
    int rowbase = b * N_ + it * 16;
#pragma unroll
    for (int d = 0; d < 8; ++d) {
        int m = d + 8 * half;
        out1[(rowbase + m) * O_ + mn] = sigmoidf_(acc[d] + bv);
    }
}

// out2: rows = (b,i,j) flat (16384), K=64: [g2(b,i,j) | g2(b,j,i)],
// g2 = [x1[b,s1,:], x2[b,s1,s2,:], red3[b,s1,s2,:]]
__global__ void out2_kernel(const float* __restrict__ x1, const float* __restrict__ x2,
                            const float* __restrict__ red3,
                            const float* __restrict__ W2, const float* __restrict__ bias2,
                            float* __restrict__ out2) {
    __shared__ float sW[64 * O_ + O_];
    for (int idx = threadIdx.x; idx < 64 * O_ + O_; idx += blockDim.x)
        sW[idx] = (idx < 64 * O_) ? W2[idx] : bias2[idx - 64 * O_];
    __syncthreads();

    int wave = (int)(blockIdx.x * blockDim.x + threadIdx.x) >> 5;  // 0..1023
    int lane = threadIdx.x & 31;
    int half = lane >> 4;
    int mn   = lane & 15;
    int jt = wave & 1;
    int i  = (wave >> 1) & (N_ - 1);
    int b  = wave >> 6;
    int j  = jt * 16 + mn;      // this lane's A-row j
    v8f acc = {0.f,0.f,0.f,0.f,0.f,0.f,0.f,0.f};
#pragma unroll
    for (int kk = 0; kk < 16; ++kk) {
        int q0 = 4 * kk + (half ? 2 : 0);
        v2f a, bm;
#pragma unroll
        for (int e = 0; e < 2; ++e) {
            int q  = q0 + e;
            int s1 = (q < 32) ? i : j;
            int s2 = (q < 32) ? j : i;
            int c  = q & 31;
            float av;
            if (c < 8)       av = x1[(b * N_ + s1) * C_ + c];
            else if (c < 16) av = x2[((b * N_ + s1) * N_ + s2) * C_ + (c - 8)];
            else             av = red3[((b * N_ + s1) * N_ + s2) * 16 + (c - 16)];
            a[e]  = av;
            bm[e] = sW[q * O_ + mn];
        }
        acc = wmma_k4(a, bm, acc);
    }
    float bv = sW[64 * O_ + mn];
    int rowbase = (b * N_ + i) * N_ + jt * 16;
#pragma unroll
    for (int d = 0; d < 8; ++d) {
        int m = d + 8 * half;
        out2[(rowbase + m) * O_ + mn] = sigmoidf_(acc[d] + bv);
    }
}

// out3: rows = (b,i,j,k) flat (524288), K=96: six perms of
// g3(b,s1,s2,s3) = [x2[b,s1,s2,:], x3[b,s1,s2,s3,:]]
__global__ void out3_kernel(const float* __restrict__ x2, const float* __restrict__ x3,
                            const float* __restrict__ W3, const float* __restrict__ bias3,
                            float* __restrict__ out3) {
    __shared__ float sW[96 * O_ + O_];
    for (int idx = threadIdx.x; idx < 96 * O_ + O_; idx += blockDim.x)
        sW[idx] = (idx < 96 * O_) ? W3[idx] : bias3[idx - 96 * O_];
    __syncthreads();

    int wave = (int)(blockIdx.x * blockDim.x + threadIdx.x) >> 5;  // 0..32767
    int lane = threadIdx.x & 31;
    int half = lane >> 4;
    int mn   = lane & 15;
    int kt = wave & 1;
    int j  = (wave >> 1) & (N_ - 1);
    int i  = (wave >> 6) & (N_ - 1);
    int b  = wave >> 11;
    int k  = kt * 16 + mn;      // this lane's A-row k
    v8f acc = {0.f,0.f,0.f,0.f,0.f,0.f,0.f,0.f};
#pragma unroll
    for (int kk = 0; kk < 24; ++kk) {
        int q0 = 4 * kk + (half ? 2 : 0);
        v2f a, bm;
#pragma unroll
        for (int e = 0; e < 2; ++e) {
            int q = q0 + e;
            int perm = q >> 4;      // folds at compile time after unroll
            int c = q & 15;
            int s1, s2, s3;
            switch (perm) {         // jnp.transpose(x,(0,)+p+(4,)) index maps
                case 0:  s1 = i; s2 = j; s3 = k; break;  // p=(1,2,3)
                case 1:  s1 = i; s2 = k; s3 = j; break;  // p=(1,3,2)
                case 2:  s1 = j; s2 = i; s3 = k; break;  // p=(2,1,3)
                case 3:  s1 = k; s2 = i; s3 = j; break;  // p=(2,3,1)
                case 4:  s1 = j; s2 = k; s3 = i; break;  // p=(3,1,2)
                default: s1 = k; s2 = j; s3 = i; break;  // p=(3,2,1)
            }
            float av;
            if (c < 8) av = x2[((b * N_ + s1) * N_ + s2) * C_ + c];
            else       av = x3[(((b * N_ + s1) * N_ + s2) * N_ + s3) * C_ + (c - 8)];
            a[e]  = av;
            bm[e] = sW[q * O_ + mn];
        }
        acc = wmma_k4(a, bm, acc);
    }
    float bv = sW[96 * O_ + mn];
    int rowbase = ((b * N_ + i) * N_ + j) * N_ + kt * 16;
#pragma unroll
    for (int d = 0; d < 8; ++d) {
        int m = d + 8 * half;
        out3[(rowbase + m) * O_ + mn] = sigmoidf_(acc[d] + bv);
    }
}

// ---------------- launcher ----------------

extern "C" void kernel_launch(void* const* d_in, const int* in_sizes, int n_in,
                              void* d_out, int out_size, void* d_ws, size_t ws_size,
                              hipStream_t stream) {
    const float* x0 = (const float*)d_in[0];
    const float* x1 = (const float*)d_in[1];
    const float* x2 = (const float*)d_in[2];
    const float* x3 = (const float*)d_in[3];
    const float* W0 = (const float*)d_in[4];
    const float* b0 = (const float*)d_in[5];
    const float* W1 = (const float*)d_in[6];
    const float* b1 = (const float*)d_in[7];
    const float* W2 = (const float*)d_in[8];
    const float* b2 = (const float*)d_in[9];
    const float* W3 = (const float*)d_in[10];
    const float* b3 = (const float*)d_in[11];

    float* ws   = (float*)d_ws;
    float* red3 = ws;                     // 262144 floats
    float* red2 = ws + 262144;            //   8192 floats
    float* red1 = ws + 262144 + 8192;     //    256 floats

    float* out0 = (float*)d_out;          //    256 floats
    float* out1 = out0 + 256;             //   8192 floats
    float* out2 = out1 + 8192;            // 262144 floats
    float* out3 = out2 + 262144;          // 8388608 floats

    // reductions first (same stream => ordered before consumers)
    red1_kernel<<<dim3(1),    dim3(256), 0, stream>>>(x1, red1);
    red2_kernel<<<dim3(32),   dim3(256), 0, stream>>>(x2, red2);
    red3_kernel<<<dim3(1024), dim3(256), 0, stream>>>(x3, red3);

    // WMMA GEMM heads
    out0_kernel<<<dim3(1),    dim3(32),  0, stream>>>(x0, red1, W0, b0, out0);
    out1_kernel<<<dim3(4),    dim3(256), 0, stream>>>(x0, x1, red2, W1, b1, out1);
    out2_kernel<<<dim3(128),  dim3(256), 0, stream>>>(x1, x2, red3, W2, b2, out2);
    out3_kernel<<<dim3(4096), dim3(256), 0, stream>>>(x2, x3, W3, b3, out3);
}